// RotatEDGNNLayer_1211180777853
// MI455X (gfx1250) — compile-verified
//
#include <hip/hip_runtime.h>
#include <stdint.h>

#define REL_DIM   128
#define ENT_DIM   256
#define LDS_TAILS 512          // LDS-accumulated tail ids: [0, 512); others fall back to global atomics
#define SLICES    4            // 4 column-slices of 32 complex components each
#define COMP_PER_SLICE 32      // complex components per slice (1 per lane)
#define WG_PER_SLICE   64      // workgroups partitioning the edge list per slice
#define THREADS   1024         // 32 waves per WG

static __device__ __forceinline__ void gatomic_add(float* p, float v) {
    unsafeAtomicAdd(p, v);     // -> global_atomic_add_f32 (coarse-grained HBM, fine here)
}

// ---------------- zero fill ----------------
__global__ void zero4_kernel(float4* __restrict__ p, long n4) {
    long i = (long)blockIdx.x * blockDim.x + threadIdx.x;
    if (i < n4) p[i] = make_float4(0.f, 0.f, 0.f, 0.f);
}

// ---------------- main: gather -> rotate -> LDS-aggregated scatter-sum ----------------
__global__ __launch_bounds__(THREADS, 1)
void rotate_scatter_kernel(const float* __restrict__ entities,
                           const float* __restrict__ relations,
                           const int*  __restrict__ eh,
                           const int*  __restrict__ er,
                           const int*  __restrict__ et,
                           float* __restrict__ outSum,   // [nEnt, 256]
                           float* __restrict__ cntg,     // [nEnt]
                           int nEdges, int nRel)
{
    // LDS: 64 KB rel slice + 128 KB accumulator + 2 KB counts = 194 KB (<= 320 KB WGP LDS)
    __shared__ float s_rel[LDS_TAILS * COMP_PER_SLICE];   // staged relation columns for this slice
    __shared__ float s_acc[LDS_TAILS * 2 * COMP_PER_SLICE]; // per-tail [re(32) | im(32)]
    __shared__ float s_cnt[LDS_TAILS];

    const int slice   = (int)blockIdx.x & (SLICES - 1);
    const int wgInSl  = (int)blockIdx.x >> 2;             // 0..WG_PER_SLICE-1
    const int tid     = (int)threadIdx.x;
    const int lane    = tid & 31;
    const int waveWg  = tid >> 5;                         // 0..31
    const int colBase = slice * COMP_PER_SLICE;           // first complex component of this slice
    const int ldsRows = (nRel < LDS_TAILS) ? nRel : LDS_TAILS;

    // ---- zero LDS accumulators ----
    for (int i = tid; i < LDS_TAILS * 2 * COMP_PER_SLICE; i += THREADS) s_acc[i] = 0.f;
    for (int i = tid; i < LDS_TAILS; i += THREADS) s_cnt[i] = 0.f;

    // ---- async-stage this slice's relation columns into LDS (CDNA5 ASYNCcnt path) ----
    {
        // per row: 32 floats = 128 B = 8 x b128 chunks; LDS layout is contiguous (idx*16)
        const int nChunks = ldsRows * 8;
        const uint32_t ldsBase = (uint32_t)(uintptr_t)(&s_rel[0]);
        const uint64_t gBase   = (uint64_t)(uintptr_t)relations + (uint64_t)colBase * 4u;
        for (int idx = tid; idx < nChunks; idx += THREADS) {
            const int row = idx >> 3, chunk = idx & 7;
            uint32_t la = ldsBase + (uint32_t)idx * 16u;
            uint64_t ga = gBase + (uint64_t)row * (REL_DIM * 4) + (uint64_t)chunk * 16u;
            asm volatile("global_load_async_to_lds_b128 %0, %1, off"
                         :: "v"(la), "v"(ga) : "memory");
        }
        asm volatile("s_wait_asynccnt 0" ::: "memory");
    }
    __syncthreads();

    const float PI_F = 3.1415926235897933f;  // reference's PI, EMB_RANGE = 1

    // ---- edge loop: this slice's WGs partition all edges; 32-edge groups per wave ----
    const int sliceWave  = wgInSl * (THREADS / 32) + waveWg;         // 0..2047 within slice
    const int wavesSlice = WG_PER_SLICE * (THREADS / 32);
    for (int base = sliceWave * 32; base < nEdges; base += wavesSlice * 32) {
        const int  e = base + lane;
        const bool valid = (e < nEdges);
        int h = 0, r = 0, t = 0;
        if (valid) { h = eh[e]; r = er[e]; t = et[e]; }
        if (slice == 0 && valid) {                                    // count once per edge
            if (t < LDS_TAILS) atomicAdd(&s_cnt[t], 1.0f);
            else               gatomic_add(&cntg[t], 1.0f);
        }
        const int n = (nEdges - base < 32) ? (nEdges - base) : 32;
        for (int j = 0; j < n; ++j) {
            const int hj = __shfl(h, j, 32);
            const int rj = __shfl(r, j, 32);
            const int tj = __shfl(t, j, 32);

            const int comp = colBase + lane;                          // complex component index
            const float rv = (rj < ldsRows) ? s_rel[rj * COMP_PER_SLICE + lane]
                                            : relations[(size_t)rj * REL_DIM + comp];
            const float re = entities[(size_t)hj * ENT_DIM + comp];
            const float im = entities[(size_t)hj * ENT_DIM + REL_DIM + comp];

            float sn, cs;
            __sincosf(rv * PI_F, &sn, &cs);
            const float re_s = re * cs - im * sn;
            const float im_s = re * sn + im * cs;

            if (tj < LDS_TAILS) {                                     // wave-uniform branch
                atomicAdd(&s_acc[tj * (2 * COMP_PER_SLICE) + lane], re_s);                 // ds_add_f32
                atomicAdd(&s_acc[tj * (2 * COMP_PER_SLICE) + COMP_PER_SLICE + lane], im_s);
            } else {
                float* orow = outSum + (size_t)tj * ENT_DIM;
                gatomic_add(&orow[comp], re_s);
                gatomic_add(&orow[REL_DIM + comp], im_s);
            }
        }
    }
    __syncthreads();

    // ---- flush LDS accumulators to global (skip untouched/zero entries) ----
    for (int idx = tid; idx < LDS_TAILS * 2 * COMP_PER_SLICE; idx += THREADS) {
        const float v = s_acc[idx];
        if (v != 0.f) {
            const int t = idx >> 6;              // /(2*COMP_PER_SLICE)
            const int c = idx & 63;
            const int col = (c < COMP_PER_SLICE) ? (colBase + c)
                                                 : (REL_DIM + colBase + (c - COMP_PER_SLICE));
            gatomic_add(&outSum[(size_t)t * ENT_DIM + col], v);
        }
    }
    if (slice == 0) {
        for (int idx = tid; idx < LDS_TAILS; idx += THREADS) {
            const float v = s_cnt[idx];
            if (v != 0.f) gatomic_add(&cntg[idx], v);
        }
    }
}

// ---------------- finalize: out[row,:] /= max(cnt[row], 1) ----------------
__global__ void finalize_kernel(float4* __restrict__ out4, const float* __restrict__ cnt, long n4) {
    long i = (long)blockIdx.x * blockDim.x + threadIdx.x;
    if (i >= n4) return;
    long row = i >> 6;                      // 64 float4 per 256-wide row
    float c = cnt[row];
    c = (c > 1.0f) ? c : 1.0f;
    const float inv = 1.0f / c;
    float4 v = out4[i];
    v.x *= inv; v.y *= inv; v.z *= inv; v.w *= inv;
    out4[i] = v;
}

extern "C" void kernel_launch(void* const* d_in, const int* in_sizes, int n_in,
                              void* d_out, int out_size, void* d_ws, size_t ws_size,
                              hipStream_t stream) {
    const float* entities  = (const float*)d_in[0];
    const float* relations = (const float*)d_in[1];
    const int*   edges     = (const int*)d_in[2];     // [3, E] int32 per harness contract
    float* out = (float*)d_out;
    float* cnt = (float*)d_ws;                        // nEnt floats of scratch

    const int nEnt   = in_sizes[0] / ENT_DIM;         // 100000
    const int nRel   = in_sizes[1] / REL_DIM;         // 500
    const int nEdges = in_sizes[2] / 3;               // 1000000

    const int* eh = edges;
    const int* er = edges + nEdges;
    const int* et = edges + 2 * (size_t)nEdges;

    // 1) zero accumulators (d_out) and counts (d_ws) every call
    const long n4o = (long)nEnt * (ENT_DIM / 4);
    const long n4c = (long)nEnt / 4;                  // nEnt divisible by 4
    zero4_kernel<<<(int)((n4o + 255) / 256), 256, 0, stream>>>((float4*)out, n4o);
    zero4_kernel<<<(int)((n4c + 255) / 256), 256, 0, stream>>>((float4*)cnt, n4c);

    // 2) LDS-aggregated gather/rotate/scatter: 4 slices x 64 WGs, 1024 threads each
    rotate_scatter_kernel<<<SLICES * WG_PER_SLICE, THREADS, 0, stream>>>(
        entities, relations, eh, er, et, out, cnt, nEdges, nRel);

    // 3) divide by clamped counts
    finalize_kernel<<<(int)((n4o + 255) / 256), 256, 0, stream>>>((float4*)out, cnt, n4o);
}